// GCNEncoder_13950053778090
// MI455X (gfx1250) — compile-verified
//
#include <hip/hip_runtime.h>

typedef __attribute__((ext_vector_type(2))) float v2f;
typedef __attribute__((ext_vector_type(8))) float v8f;

#define GCN_IN_C   128
#define GCN_HID_C  128
#define GCN_OUT_C  64
#define GCN_BN_EPS 1e-5f

// ---------------------------------------------------------------- utilities

__global__ void zero_f32(float* __restrict__ p, long n) {
  long i = (long)blockIdx.x * blockDim.x + threadIdx.x;
  long stride = (long)gridDim.x * blockDim.x;
  for (; i < n; i += stride) p[i] = 0.0f;
}

__global__ void degree_kernel(const int* __restrict__ dst, int E, int* __restrict__ deg) {
  int i = blockIdx.x * blockDim.x + threadIdx.x;
  if (i < E) atomicAdd(&deg[dst[i]], 1);
}

__global__ void dinv_kernel(const int* __restrict__ deg, float* __restrict__ dinv, int N) {
  int i = blockIdx.x * blockDim.x + threadIdx.x;
  if (i < N) dinv[i] = rsqrtf((float)(deg[i] + 1));  // +1 self-loop; always > 0
}

// ----------------------------------------------------- fp32 WMMA dense GEMM
// C[M,N] = A[M,KDIM] @ B[KDIM,N].
// One block = (N/16) waves = one 16-row output stripe.  The 16 x KDIM A tile
// is staged once into LDS (padded row stride KDIM+4 -> lanes 0..15 read banks
// 4*l16, conflict-free ds_load_b64), then each wave owns one 16x16 N-tile and
// runs the K loop on V_WMMA_F32_16X16X4_F32.
// Operand layout (ISA 7.12.2):
//   A 16x4:  lanes 0-15 -> M=lane, K={k0,k0+1}; lanes 16-31 -> K={k0+2,k0+3}
//   B 4x16:  mirrored over N (lane%16 = N, lane/16 selects K pair)
//   C 16x16: VGPR r: lanes 0-15 -> (M=r, N=lane); lanes 16-31 -> (M=r+8, N=lane-16)
template <int KDIM>
__global__ void gemm_wmma_lds(const float* __restrict__ A, const float* __restrict__ B,
                              float* __restrict__ C, int N) {
  constexpr int LDA = KDIM + 4;                 // LDS row pad (floats)
  __shared__ float As[16 * LDA];

  const int tm = blockIdx.x;                    // 16-row stripe index
  const float* __restrict__ gA = A + (long)tm * 16 * KDIM;

  // cooperative stage: 16*KDIM floats as float4s
  for (int idx = threadIdx.x; idx < 16 * KDIM / 4; idx += blockDim.x) {
    const int r  = idx / (KDIM / 4);
    const int c4 = idx % (KDIM / 4);
    const float4 v = ((const float4*)(gA + (long)r * KDIM))[c4];
    *(float4*)&As[r * LDA + c4 * 4] = v;
  }
  __syncthreads();

  const int lane = threadIdx.x & 31;
  const int half = lane >> 4;                   // 0 or 1
  const int l16  = lane & 15;
  const int tn   = threadIdx.x >> 5;            // this wave's N tile
  const int bcol = tn * 16 + l16;
  const float* __restrict__ ap = &As[l16 * LDA];

  v8f acc = {};
#pragma unroll 4
  for (int k0 = 0; k0 < KDIM; k0 += 4) {
    const int ka = k0 + 2 * half;
    v2f a, b;
    a.x = ap[ka];                               // ds_load_b64 (contiguous pair)
    a.y = ap[ka + 1];
    b.x = B[(long)ka * N + bcol];
    b.y = B[(long)(ka + 1) * N + bcol];
    acc = __builtin_amdgcn_wmma_f32_16x16x4_f32(
        /*neg_a=*/false, a, /*neg_b=*/false, b,
        /*c_mod=*/(short)0, acc, /*reuse_a=*/false, /*reuse_b=*/false);
  }

  const int orow = tm * 16 + half * 8;
  float* __restrict__ cp = C + (long)orow * N + bcol;
#pragma unroll
  for (int r = 0; r < 8; ++r) cp[(long)r * N] = acc[r];
}

// ------------------------------------------------- edge scatter (msg passing)
// One edge per wave: wave-uniform src/dst, coalesced 512B gather via b128,
// per-dword global_atomic_add_f32 scatter (accumulators are L2-resident).
__global__ void scatter_c128(const float* __restrict__ h, const int* __restrict__ src,
                             const int* __restrict__ dst, const float* __restrict__ dinv,
                             float* __restrict__ agg, int E) {
  const int wave = (int)((blockIdx.x * (long)blockDim.x + threadIdx.x) >> 5);
  const int lane = threadIdx.x & 31;
  if (wave >= E) return;
  const int s = src[wave];
  const int d = dst[wave];
  const float coef = dinv[s] * dinv[d];
  const float4 v = ((const float4*)(h + (long)s * 128))[lane];
  float* __restrict__ o = agg + (long)d * 128 + lane * 4;
  atomicAdd(o + 0, v.x * coef);
  atomicAdd(o + 1, v.y * coef);
  atomicAdd(o + 2, v.z * coef);
  atomicAdd(o + 3, v.w * coef);
}

__global__ void scatter_c64(const float* __restrict__ h, const int* __restrict__ src,
                            const int* __restrict__ dst, const float* __restrict__ dinv,
                            float* __restrict__ agg, int E) {
  const int wave = (int)((blockIdx.x * (long)blockDim.x + threadIdx.x) >> 5);
  const int lane = threadIdx.x & 31;
  if (wave >= E) return;
  const int s = src[wave];
  const int d = dst[wave];
  const float coef = dinv[s] * dinv[d];
  const float2 v = ((const float2*)(h + (long)s * 64))[lane];
  float* __restrict__ o = agg + (long)d * 64 + lane * 2;
  atomicAdd(o + 0, v.x * coef);
  atomicAdd(o + 1, v.y * coef);
}

// ------------------------------------------- finalize: self-loop + bias + BN
__global__ void finalize_relu_c128(const float* __restrict__ h, float* __restrict__ agg,
                                   const float* __restrict__ dinv, const float* __restrict__ bias,
                                   const float* __restrict__ gamma, const float* __restrict__ beta,
                                   const float* __restrict__ mean, const float* __restrict__ var,
                                   int N) {
  const long idx = (long)blockIdx.x * blockDim.x + threadIdx.x;
  if (idx >= (long)N * 128) return;
  const int i = (int)(idx >> 7);
  const int c = (int)(idx & 127);
  const float di = dinv[i];
  float v = agg[idx] + h[idx] * di * di + bias[c];
  v = (v - mean[c]) * rsqrtf(var[c] + GCN_BN_EPS) * gamma[c] + beta[c];
  agg[idx] = fmaxf(v, 0.0f);  // in-place: agg becomes layer-1 activation
}

__global__ void finalize_c64(const float* __restrict__ h, const float* __restrict__ agg,
                             const float* __restrict__ dinv, const float* __restrict__ bias,
                             const float* __restrict__ gamma, const float* __restrict__ beta,
                             const float* __restrict__ mean, const float* __restrict__ var,
                             float* __restrict__ out, int N) {
  const long idx = (long)blockIdx.x * blockDim.x + threadIdx.x;
  if (idx >= (long)N * 64) return;
  const int i = (int)(idx >> 6);
  const int c = (int)(idx & 63);
  const float di = dinv[i];
  float v = agg[idx] + h[idx] * di * di + bias[c];
  v = (v - mean[c]) * rsqrtf(var[c] + GCN_BN_EPS) * gamma[c] + beta[c];
  out[idx] = v;
}

// ---------------------------------------------------------------- launcher

extern "C" void kernel_launch(void* const* d_in, const int* in_sizes, int n_in,
                              void* d_out, int out_size, void* d_ws, size_t ws_size,
                              hipStream_t stream) {
  const float* x   = (const float*)d_in[0];
  const int*   ei  = (const int*)d_in[1];
  const float* W1  = (const float*)d_in[2];
  const float* b1  = (const float*)d_in[3];
  const float* g1  = (const float*)d_in[4];
  const float* be1 = (const float*)d_in[5];
  const float* m1  = (const float*)d_in[6];
  const float* v1  = (const float*)d_in[7];
  const float* W2  = (const float*)d_in[8];
  const float* b2  = (const float*)d_in[9];
  const float* g2  = (const float*)d_in[10];
  const float* be2 = (const float*)d_in[11];
  const float* m2  = (const float*)d_in[12];
  const float* v2  = (const float*)d_in[13];

  const int N = in_sizes[0] / GCN_IN_C;   // 50000
  const int E = in_sizes[1] / 2;          // 1,600,000
  const int* src = ei;
  const int* dst = ei + E;

  // workspace carve-out (16B-aligned sizes)
  char* ws = (char*)d_ws;
  int*   deg  = (int*)ws;                         ws += (size_t)N * sizeof(int);
  float* dinv = (float*)ws;                       ws += (size_t)N * sizeof(float);
  float* h1   = (float*)ws;                       ws += (size_t)N * GCN_HID_C * sizeof(float);
  float* agg1 = (float*)ws;                       ws += (size_t)N * GCN_HID_C * sizeof(float);
  float* h2   = (float*)ws;                       ws += (size_t)N * GCN_OUT_C * sizeof(float);
  float* agg2 = (float*)ws;                       ws += (size_t)N * GCN_OUT_C * sizeof(float);
  (void)ws_size; (void)n_in; (void)out_size;

  const int ZB = 2048;
  // zero accumulators + degree counts (poisoned workspace)
  zero_f32<<<ZB, 256, 0, stream>>>((float*)deg, (long)N);
  zero_f32<<<ZB, 256, 0, stream>>>(agg1, (long)N * GCN_HID_C);
  zero_f32<<<ZB, 256, 0, stream>>>(agg2, (long)N * GCN_OUT_C);

  // degrees + dinv (shared by both layers: same graph)
  degree_kernel<<<(E + 255) / 256, 256, 0, stream>>>(dst, E, deg);
  dinv_kernel<<<(N + 255) / 256, 256, 0, stream>>>(deg, dinv, N);

  // ---- layer 1: h1 = x @ W1 (WMMA from LDS), aggregate, BN+ReLU
  gemm_wmma_lds<GCN_IN_C><<<N / 16, (GCN_HID_C / 16) * 32, 0, stream>>>(x, W1, h1, GCN_HID_C);
  {
    const long blocks = ((long)E * 32 + 255) / 256;
    scatter_c128<<<(int)blocks, 256, 0, stream>>>(h1, src, dst, dinv, agg1, E);
  }
  {
    const long total = (long)N * GCN_HID_C;
    finalize_relu_c128<<<(int)((total + 255) / 256), 256, 0, stream>>>(
        h1, agg1, dinv, b1, g1, be1, m1, v1, N);
  }

  // ---- layer 2: h2 = act1 @ W2 (WMMA from LDS), aggregate, BN
  gemm_wmma_lds<GCN_HID_C><<<N / 16, (GCN_OUT_C / 16) * 32, 0, stream>>>(agg1, W2, h2, GCN_OUT_C);
  {
    const long blocks = ((long)E * 32 + 255) / 256;
    scatter_c64<<<(int)blocks, 256, 0, stream>>>(h2, src, dst, dinv, agg2, E);
  }
  {
    const long total = (long)N * GCN_OUT_C;
    finalize_c64<<<(int)((total + 255) / 256), 256, 0, stream>>>(
        h2, agg2, dinv, b2, g2, be2, m2, v2, (float*)d_out, N);
  }
}